// PR_53420803227933
// MI455X (gfx1250) — compile-verified
//
#include <hip/hip_runtime.h>
#include <hip/hip_bf16.h>

typedef float v2f __attribute__((ext_vector_type(2)));
typedef float v8f __attribute__((ext_vector_type(8)));

#define T_LEN   8192
#define LOG2T   13
#define NB      8      // batch
#define NC      16     // channels in/out
#define NM      16     // modes

__device__ __forceinline__ int bitrev13(int p) {
    return (int)(__brev((unsigned)p) >> 19);
}

// ---------------------------------------------------------------------------
// Kernel 0: zero the res2 accumulator region (8*16*16 complex = 4096 floats)
// ---------------------------------------------------------------------------
__global__ void k_zero_res2(float* res2) {
    int i = blockIdx.x * blockDim.x + threadIdx.x;
    if (i < NB * NC * NM * 2) res2[i] = 0.0f;
}

// ---------------------------------------------------------------------------
// Kernel 1: forward FFT, radix-2 DIF in LDS. natural input -> bit-reversed
// output order. One block per (b,i) series; 8192 complex floats = 64KB LDS.
// ---------------------------------------------------------------------------
__global__ void k_fft_fwd(const float* __restrict__ x, float2* __restrict__ alpha) {
    __shared__ float2 S[T_LEN];
    const int bi = blockIdx.x;
    const float* xp = x + ((size_t)bi << LOG2T);
    for (int j = threadIdx.x; j < T_LEN; j += blockDim.x)
        S[j] = make_float2(xp[j], 0.0f);
    __syncthreads();
    for (int s = 0; s < LOG2T; ++s) {
        const int m = T_LEN >> s, half = m >> 1;
        const float w = -6.283185307179586f / (float)m;
        for (int q = threadIdx.x; q < (T_LEN >> 1); q += blockDim.x) {
            const int j  = q >> (12 - s);
            const int tt = q & (half - 1);
            const int pos = (j << (LOG2T - s)) + tt;
            float2 u = S[pos], v = S[pos + half];
            float sn, cs; __sincosf(w * (float)tt, &sn, &cs);
            float dx = u.x - v.x, dy = u.y - v.y;
            S[pos]        = make_float2(u.x + v.x, u.y + v.y);
            S[pos + half] = make_float2(dx * cs - dy * sn, dx * sn + dy * cs);
        }
        __syncthreads();
    }
    float2* ap = alpha + ((size_t)bi << LOG2T);
    for (int j = threadIdx.x; j < T_LEN; j += blockDim.x) ap[j] = S[j];
}

// ---------------------------------------------------------------------------
// Kernel 2: per-frequency rational filter + matvec + res2 reduction.
// Storage position p holds frequency f = bitrev13(p).
// 256 threads; thread t owns (o = t>>4, k = t&15); loops i = 0..15.
// ---------------------------------------------------------------------------
__global__ void k_mid(const float2* __restrict__ alpha,
                      const float* __restrict__ wp_re, const float* __restrict__ wp_im,
                      const float* __restrict__ wr_re, const float* __restrict__ wr_im,
                      const float* __restrict__ grid,
                      float2* __restrict__ res1, float* __restrict__ res2,
                      int p_per_block) {
    __shared__ float2 aS[NC * NB];       // alpha[i][b] for current p      (1 KB)
    __shared__ float2 hS[NC * 256];      // H[i][(o,k)=t]                 (32 KB)
    __shared__ float2 hSumS[NC * NC];    // sum_k H[i][o]                  (2 KB)
    const int t = threadIdx.x;
    const int o = t >> 4, k = t & 15;
    const float dt = grid[1] - grid[0];

    // per-thread parameters for the 16 i values (kept in registers)
    float pr[NC], pi[NC], rr[NC], ri[NC];
#pragma unroll
    for (int i = 0; i < NC; ++i) {
        const int idx = (i * NC + o) * NM + k;
        pr[i] = wp_re[idx]; pi[i] = wp_im[idx];
        rr[i] = wr_re[idx]; ri[i] = wr_im[idx];
    }
    float r2re[NB], r2im[NB];
#pragma unroll
    for (int b = 0; b < NB; ++b) { r2re[b] = 0.0f; r2im[b] = 0.0f; }

    const int p0 = blockIdx.x * p_per_block;
    for (int pp = 0; pp < p_per_block; ++pp) {
        const int p = p0 + pp;
        const int f = bitrev13(p);
        const float fr  = (f < (T_LEN >> 1)) ? (float)f : (float)(f - T_LEN);
        const float lam = 6.283185307179586f * fr / ((float)T_LEN * dt);

        __syncthreads();                 // previous iteration fully consumed
        if (t < NC * NB) {
            const int i = t >> 3, b = t & 7;
            aS[i * NB + b] = alpha[(((size_t)(b * NC + i)) << LOG2T) + p];
        }
        __syncthreads();

        // Phase A: H = wr / (i*lam - wp); stage in LDS; accumulate res2.
#pragma unroll
        for (int i = 0; i < NC; ++i) {
            const float dr = -pr[i], di = lam - pi[i];
            const float inv = __frcp_rn(dr * dr + di * di);
            const float hre = (rr[i] * dr + ri[i] * di) * inv;
            const float him = (ri[i] * dr - rr[i] * di) * inv;
            hS[i * 256 + t] = make_float2(hre, him);
#pragma unroll
            for (int b = 0; b < NB; ++b) {
                const float2 a = aS[i * NB + b];
                r2re[b] -= a.x * hre - a.y * him;   // res2 = -sum alpha*H
                r2im[b] -= a.x * him + a.y * hre;
            }
        }
        __syncthreads();

        // Phase A.5: cooperative Hsum[i][o] = sum_k H[i][o][k] (one (i,o)/thread)
        {
            const int i = t >> 4, oo = t & 15;
            float hsre = 0.0f, hsim = 0.0f;
#pragma unroll
            for (int kk = 0; kk < NM; ++kk) {
                const float2 h = hS[i * 256 + oo * NM + kk];
                hsre += h.x; hsim += h.y;
            }
            hSumS[i * NC + oo] = make_float2(hsre, hsim);
        }
        __syncthreads();

        // Phase B: res1[b,o,p] = sum_i alpha[b,i] * Hsum[i,o]
        if (t < NB * NC) {
            const int b = t >> 4, oo = t & 15;
            float sre = 0.0f, sim = 0.0f;
#pragma unroll
            for (int i = 0; i < NC; ++i) {
                const float2 hs = hSumS[i * NC + oo];
                const float2 a  = aS[i * NB + b];
                sre += a.x * hs.x - a.y * hs.y;
                sim += a.x * hs.y + a.y * hs.x;
            }
            res1[(((size_t)(b * NC + oo)) << LOG2T) + p] = make_float2(sre, sim);
        }
    }

    // one atomic flush per thread per batch element
#pragma unroll
    for (int b = 0; b < NB; ++b) {
        float* dst = res2 + (((b * NC + o) * NM + k) << 1);
        atomicAdd(dst + 0, r2re[b]);
        atomicAdd(dst + 1, r2im[b]);
    }
}

// ---------------------------------------------------------------------------
// Kernel 3: inverse FFT, radix-2 DIT, bit-reversed input -> natural output.
// Writes x1 = Re(ifft) -> out (scaled by 1/T). One block per (b,o).
// ---------------------------------------------------------------------------
__global__ void k_ifft(const float2* __restrict__ res1, float* __restrict__ out) {
    __shared__ float2 S[T_LEN];
    const int bo = blockIdx.x;
    const float2* rp = res1 + ((size_t)bo << LOG2T);
    for (int j = threadIdx.x; j < T_LEN; j += blockDim.x) S[j] = rp[j];
    __syncthreads();
    for (int st = 1; st <= LOG2T; ++st) {
        const int m = 1 << st, half = m >> 1;
        const float w = 6.283185307179586f / (float)m;   // +i twiddles (inverse)
        for (int q = threadIdx.x; q < (T_LEN >> 1); q += blockDim.x) {
            const int j  = q >> (st - 1);
            const int tt = q & (half - 1);
            const int pos = j * m + tt;
            const float2 u = S[pos];
            float2 v = S[pos + half];
            float sn, cs; __sincosf(w * (float)tt, &sn, &cs);
            const float vr = v.x * cs - v.y * sn;
            const float vi = v.x * sn + v.y * cs;
            S[pos]        = make_float2(u.x + vr, u.y + vi);
            S[pos + half] = make_float2(u.x - vr, u.y - vi);
        }
        __syncthreads();
    }
    float* op = out + ((size_t)bo << LOG2T);
    const float invT = 1.0f / (float)T_LEN;
    for (int j = threadIdx.x; j < T_LEN; j += blockDim.x) op[j] = S[j].x * invT;
}

// ---------------------------------------------------------------------------
// Kernel 4: x2 via WMMA f32 16x16x4.
//   out[b,o,z] += (1/T) * Re( sum_{c,k} res2[b,c,k] * exp(wp[c,o,k]*t_z) )
// D(16x16) = A(16x4) x B(4x16) accumulated over K=512:
//   A[b, 2m]   =  res2_re[b, m]     (rows 8..15 zero-padded)
//   A[b, 2m+1] = -res2_im[b, m]
//   B[2m,   z] =  e^{a t} cos(w t)
//   B[2m+1, z] =  e^{a t} sin(w t)
// Block = 256 threads = 8 waves; blockIdx = o * 64 + zgroup; wave -> z-tile.
// ---------------------------------------------------------------------------
__global__ void k_x2_wmma(const float* __restrict__ res2,
                          const float* __restrict__ wp_re, const float* __restrict__ wp_im,
                          const float* __restrict__ grid,
                          float* __restrict__ out) {
    __shared__ float  As[512][16];     // [K][M] : sign-folded res2, 32KB
    __shared__ float2 wpS[256];        // (a,w) per (c,k) pair for this o, 2KB
    const int o    = blockIdx.x >> 6;
    const int zg   = blockIdx.x & 63;
    const int wave = threadIdx.x >> 5;
    const int lane = threadIdx.x & 31;
    const int z0   = (zg * 8 + wave) << 4;

    // stage A matrix (same for every block; cheap) and wp pairs for this o
    for (int idx = threadIdx.x; idx < 512 * 16; idx += blockDim.x) {
        const int col = idx >> 4, row = idx & 15;   // col = K index, row = M (=b)
        float v = 0.0f;
        if (row < NB) {
            const int m = col >> 1;                 // (c,k) pair
            const float* r = res2 + (((row * NC + (m >> 4)) * NM + (m & 15)) << 1);
            v = (col & 1) ? -r[1] : r[0];
        }
        As[col][row] = v;
    }
    if (threadIdx.x < 256) {
        const int m = threadIdx.x;
        const int pidx = ((m >> 4) * NC + o) * NM + (m & 15);
        wpS[m] = make_float2(wp_re[pidx], wp_im[pidx]);
    }
    __syncthreads();

    // per-lane fragment coordinates (ISA 7.12.2: 32-bit A 16x4 layout)
    const int rowA = lane & 15;            // M
    const int kb   = (lane >> 4) << 1;     // K base: 0 (lanes 0-15) or 2 (16-31)
    const int nB   = lane & 15;            // N column for B fragment
    const float tz = grid[z0 + nB];

    v8f acc = {0.f, 0.f, 0.f, 0.f, 0.f, 0.f, 0.f, 0.f};
    for (int kk = 0; kk < 128; ++kk) {
        const int k0 = (kk << 2) + kb;     // even K index handled by this lane
        v2f a = { As[k0][rowA], As[k0 + 1][rowA] };
        const int m = k0 >> 1;             // (c,k) pair
        const float2 pw = wpS[m];
        const float ex = __expf(pw.x * tz);
        float sn, cs; __sincosf(pw.y * tz, &sn, &cs);
        v2f b = { ex * cs, ex * sn };      // (E_re, E_im) rows k0, k0+1 at col nB
        acc = __builtin_amdgcn_wmma_f32_16x16x4_f32(
            false, a, false, b, (short)0, acc, false, false);
    }

    // C/D layout: VGPR r, lanes 0-15 -> M=r (b = 0..7 valid), N = lane
    if (lane < 16) {
        const float invT = 1.0f / (float)T_LEN;
        const int z = z0 + lane;
#pragma unroll
        for (int r = 0; r < NB; ++r) {
            float* dst = out + (((size_t)(r * NC + o)) << LOG2T) + z;
            *dst += acc[r] * invT;
        }
    }
}

// ---------------------------------------------------------------------------
// Host-side launcher
// ---------------------------------------------------------------------------
extern "C" void kernel_launch(void* const* d_in, const int* in_sizes, int n_in,
                              void* d_out, int out_size, void* d_ws, size_t ws_size,
                              hipStream_t stream) {
    const float* x     = (const float*)d_in[0];   // (8,16,8192)
    const float* grid  = (const float*)d_in[1];   // (8192)
    const float* wp_re = (const float*)d_in[2];   // (16,16,16)
    const float* wp_im = (const float*)d_in[3];
    const float* wr_re = (const float*)d_in[4];
    const float* wr_im = (const float*)d_in[5];
    float* out = (float*)d_out;                   // (8,16,8192)

    char* ws = (char*)d_ws;
    float2* alpha = (float2*)(ws);                          // 8 MB
    float2* res1  = (float2*)(ws + (size_t)8 * 1024 * 1024);// 8 MB
    float*  res2  = (float*)(ws + (size_t)16 * 1024 * 1024);// 16 KB

    k_zero_res2<<<16, 256, 0, stream>>>(res2);
    k_fft_fwd<<<NB * NC, 256, 0, stream>>>(x, alpha);
    // 256 blocks x 32 frequencies: fills the WGP array far better than 64 blocks
    k_mid<<<256, 256, 0, stream>>>(alpha, wp_re, wp_im, wr_re, wr_im, grid,
                                   res1, res2, T_LEN / 256);
    k_ifft<<<NB * NC, 256, 0, stream>>>(res1, out);
    k_x2_wmma<<<NC * 64, 256, 0, stream>>>(res2, wp_re, wp_im, grid, out);
}